// Embedding_19937238188585
// MI455X (gfx1250) — compile-verified
//
#include <hip/hip_runtime.h>

// Embedding gather: out[row, :] = W[ids[row], :]
//   ids: int32[16384]  (4 x 4096 flattened)
//   W:   float32[50257, 1024]
//   out: float32[16384, 1024]
// Pure bandwidth problem (~128 MiB total traffic -> ~5.5us at 23.3 TB/s).
// One 256-thread block per row; each lane moves one 16-byte chunk.

typedef __attribute__((ext_vector_type(4))) float v4f;
typedef __attribute__((ext_vector_type(4))) int   v4i;

#define EMBED 1024
#define THREADS 256   // 256 lanes * 16B = 4096B = one full embedding row

// Prefer the gfx1250 async VMEM<->LDS engine (ASYNCcnt path) when the
// toolchain exposes the builtins; otherwise fall back to a plain
// 128-bit register copy with a non-temporal store.
#if defined(__gfx1250__) && \
    __has_builtin(__builtin_amdgcn_global_load_async_to_lds_b128) && \
    __has_builtin(__builtin_amdgcn_global_store_async_from_lds_b128) && \
    __has_builtin(__builtin_amdgcn_s_wait_asynccnt)
#define USE_ASYNC_LDS 1
#else
#define USE_ASYNC_LDS 0
#endif

__global__ __launch_bounds__(THREADS) void
embed_gather_kernel(const int* __restrict__ ids,
                    const float* __restrict__ w,
                    float* __restrict__ out)
{
    const int row = blockIdx.x;          // token index, 0..16383
    const int tok = ids[row];            // block-uniform -> s_load_b32

#if USE_ASYNC_LDS
    // mem -> LDS -> mem through the async DMA path; data never touches VGPRs.
    // Each lane's LDS slot is private to its own wave, and ASYNCcnt is
    // per-wave, so s_wait_asynccnt is sufficient ordering (no barrier).
    __shared__ v4i tile[THREADS];

    v4i* gsrc = (v4i*)(w   + (size_t)tok * EMBED) + threadIdx.x;
    v4i* gdst = (v4i*)(out + (size_t)row * EMBED) + threadIdx.x;
    v4i* lptr = &tile[threadIdx.x];

    __builtin_amdgcn_global_load_async_to_lds_b128(gsrc, lptr, 0, 0);
    __builtin_amdgcn_s_wait_asynccnt(0);          // LDS tile resident
    __builtin_amdgcn_global_store_async_from_lds_b128(gdst, lptr, 0, 0);
    __builtin_amdgcn_s_wait_asynccnt(0);          // store handed to memory
#else
    // Fallback: straight 128-bit copy. Weights load stays RT (repeat tokens
    // hit the 192MB L2); output is write-once -> non-temporal store.
    const v4f* __restrict__ src = (const v4f*)(w + (size_t)tok * EMBED);
    v4f* __restrict__ dst       = (v4f*)(out + (size_t)row * EMBED);
    v4f v = src[threadIdx.x];
    __builtin_nontemporal_store(v, &dst[threadIdx.x]);
#endif
}

extern "C" void kernel_launch(void* const* d_in, const int* in_sizes, int n_in,
                              void* d_out, int out_size, void* d_ws, size_t ws_size,
                              hipStream_t stream)
{
    const int*   ids = (const int*)d_in[0];   // token_ids, int32, flat [B*S]
    const float* w   = (const float*)d_in[1]; // weights, f32 [VOCAB, EMBED]
    float*       out = (float*)d_out;         // f32 [B*S, EMBED]

    const int rows = in_sizes[0];             // 4 * 4096 = 16384
    embed_gather_kernel<<<rows, THREADS, 0, stream>>>(ids, w, out);
}